// Model_41266045780687
// MI455X (gfx1250) — compile-verified
//
#include <hip/hip_runtime.h>
#include <stdint.h>

// ---------------------------------------------------------------------------
// Fused pixel-unshuffle + channel-softmax attention + fold + sigmoid gate.
//
// Algebraic collapse: the unfold/softmax/fold pipeline reduces exactly to
//   folded[b,h,w] = Nh*Nw * sum_c v_c*softmax(v)_c
// over the 12 unshuffled channel values v of the 2x2 block at (h,w), with
// N=2 at borders else 3. Output is x * (0.9*sigmoid(folded_upsampled)+0.1).
//
// Data movement: per workgroup a 12 KB tile (3 channels x 2 contiguous rows)
// is staged into LDS by the CDNA5 Tensor Data Mover (tensor_load_to_lds,
// TENSORcnt), then 256 lanes compute one downscaled pixel each and store
// coalesced float2s directly to HBM. ~50 MB total traffic -> HBM-bound
// (~2.2 us roofline at 23.3 TB/s).
// ---------------------------------------------------------------------------

#define BATCH  8
#define CH     3
#define IMG_H  512
#define IMG_W  512
#define HS     (IMG_H / 2)
#define WS     (IMG_W / 2)
#define PLANE  (IMG_H * IMG_W)        // 262144 elements per channel plane

typedef uint32_t u32;
typedef uint64_t u64;
typedef u32 v4u __attribute__((ext_vector_type(4)));
typedef int v8i __attribute__((ext_vector_type(8)));
typedef int v4i __attribute__((ext_vector_type(4)));

__global__ __launch_bounds__(256)
void fused_pu_softmax_gate(const float* __restrict__ x, float* __restrict__ out)
{
    // [channel][row(2)][col(512)] staging tile, 12 KB
    __shared__ __align__(128) float smem[CH * 2 * IMG_W];

    const u32 bh = (u32)blockIdx.x;       // 0 .. BATCH*HS-1
    const u32 b  = bh >> 8;               // bh / HS   (HS == 256)
    const u32 h  = bh & (HS - 1);         // bh % HS   -> downscaled row
    // element offset of row pair (2h, 2h+1) of channel 0:
    //   b*CH*PLANE + 2h*IMG_W == (b*CH*IMG_H/2 + h) * (2*IMG_W)
    const size_t base = ((size_t)(b * (CH * HS) + h)) * (size_t)(2 * IMG_W);

    const u64 gaddr = (u64)(uintptr_t)(x + base);

    // ---- TDM: wave 0 issues the tensor DMA for the whole workgroup --------
    if (threadIdx.x < 32) {
        const u32 lds_off = (u32)(uintptr_t)(&smem[0]);   // low 32b of flat LDS addr

        // D# group 0: count=1 (valid), lds_addr, global_addr[56:0], type=2
        v4u g0;
        g0[0] = 1u;
        g0[1] = lds_off;
        g0[2] = (u32)gaddr;
        g0[3] = (u32)((gaddr >> 32) & 0x01FFFFFFu) | (2u << 30);

        // D# group 1: data_size=4B; tensor_dim0=PLANE, tensor_dim1=CH;
        // tile_dim0 = 2*IMG_W (two contiguous rows), tile_dim1 = CH;
        // tensor_dim0_stride = PLANE (channel stride). 2-D tile.
        v8i g1;
        g1[0] = (int)(2u << 16);                              // data_size = 4 bytes
        g1[1] = (int)((u32)(PLANE & 0xFFFF) << 16);           // tensor_dim0[15:0]
        g1[2] = (int)(((u32)PLANE >> 16) | ((u32)CH << 16));  // td0[31:16] | td1[15:0]
        g1[3] = (int)((u32)(2 * IMG_W) << 16);                // tile_dim0 = 1024
        g1[4] = (int)CH;                                      // tile_dim1=3, tile_dim2=0
        g1[5] = (int)PLANE;                                   // tensor_dim0_stride lo32
        g1[6] = 0;                                            // stride hi, dim1_stride lo
        g1[7] = 0;

        v4i zero4 = {0, 0, 0, 0};                             // groups 2/3: 2-D tile
        v8i zero8 = {0, 0, 0, 0, 0, 0, 0, 0};
        __builtin_amdgcn_tensor_load_to_lds(g0, g1, zero4, zero4, zero8, 0);
        __builtin_amdgcn_s_wait_tensorcnt(0);
    }
    __syncthreads();

    // ---- per-lane compute: one downscaled pixel ---------------------------
    const u32 w = threadIdx.x;            // downscaled column, 0..255

    float v[12];
#pragma unroll
    for (int c = 0; c < CH; ++c) {
        const float2 r0 = *(const float2*)&smem[c * (2 * IMG_W) + 2 * w];
        const float2 r1 = *(const float2*)&smem[c * (2 * IMG_W) + IMG_W + 2 * w];
        v[c * 4 + 0] = r0.x;   // (i=0, j=0)
        v[c * 4 + 1] = r0.y;   // (i=0, j=1)
        v[c * 4 + 2] = r1.x;   // (i=1, j=0)
        v[c * 4 + 3] = r1.y;   // (i=1, j=1)
    }

    float m = v[0];
#pragma unroll
    for (int k = 1; k < 12; ++k) m = fmaxf(m, v[k]);

    float s = 0.0f, gs = 0.0f;
#pragma unroll
    for (int k = 0; k < 12; ++k) {
        const float e = __expf(v[k] - m);
        s += e;
        gs = fmaf(v[k], e, gs);
    }
    const float g = gs * __builtin_amdgcn_rcpf(s);

    const int   nh = (h == 0u || h == HS - 1) ? 2 : 3;
    const int   nw = (w == 0u || w == WS - 1) ? 2 : 3;
    const float f  = (float)(nh * nw) * g;

    const float sig = __builtin_amdgcn_rcpf(1.0f + __expf(-f));
    const float a   = fmaf(0.9f, sig, 0.1f);

    float* dst = out + base + 2 * w;
#pragma unroll
    for (int c = 0; c < CH; ++c) {
        float2 o0, o1;
        o0.x = v[c * 4 + 0] * a;  o0.y = v[c * 4 + 1] * a;
        o1.x = v[c * 4 + 2] * a;  o1.y = v[c * 4 + 3] * a;
        *(float2*)(dst + (size_t)c * PLANE)         = o0;
        *(float2*)(dst + (size_t)c * PLANE + IMG_W) = o1;
    }
}

extern "C" void kernel_launch(void* const* d_in, const int* in_sizes, int n_in,
                              void* d_out, int out_size, void* d_ws, size_t ws_size,
                              hipStream_t stream)
{
    (void)in_sizes; (void)n_in; (void)d_ws; (void)ws_size; (void)out_size;
    const float* x   = (const float*)d_in[0];
    float*       out = (float*)d_out;

    const int blocks = BATCH * HS;        // 2048 workgroups, one per (b, h)
    fused_pu_softmax_gate<<<blocks, 256, 0, stream>>>(x, out);
}